// ICP_3959959847378
// MI455X (gfx1250) — compile-verified
//
#include <hip/hip_runtime.h>

#define H_IMG 1024
#define W_IMG 1280
#define NPIX (H_IMG * W_IMG)
#define NORM_THR 0.93969262078590838405f /* cos(20 deg) */
#define DIST_THR2 0.04f                  /* 0.2^2 */
#define DAMPING 1e-6f

typedef float v2f __attribute__((ext_vector_type(2)));
typedef float v8f __attribute__((ext_vector_type(8)));

// ---------------------------------------------------------------------------
// Copy initial pose (d_in[0]) into workspace pose slot.
// ---------------------------------------------------------------------------
__global__ void __launch_bounds__(32) icp_init_pose(const float* __restrict__ pose_in,
                                                    float* __restrict__ pose_ws) {
  if (threadIdx.x < 16) pose_ws[threadIdx.x] = pose_in[threadIdx.x];
}

// ---------------------------------------------------------------------------
// Map + WMMA Gram reduction.
// Per pixel builds G = [cross(p,rn1), rn1, res, valid] (zeroed when invalid).
// Each wave stages 32 pixels x 8 floats of G in its private LDS slice, then
// accumulates G^T G with 8x V_WMMA_F32_16X16X4_F32 (K = 4 pixels per wmma).
// For a Gram product the CDNA5 A(16x4) and B(4x16) lane layouts coincide, so
// the same VGPR pair feeds both A and B.
// ---------------------------------------------------------------------------
__global__ void __launch_bounds__(256) icp_reduce(
    const float* __restrict__ v0g, const float* __restrict__ v1g,
    const float* __restrict__ n0g, const float* __restrict__ n1g,
    const float* __restrict__ Kg,  const float* __restrict__ pose,
    float* __restrict__ partials /* gridDim.x * 64 floats */) {
  __shared__ float sG[8][256];  // per-wave staging: 32 pixels * 8 comps
  __shared__ float sAcc[8][64]; // per-wave 8x8 Gram partials

  const int tid  = threadIdx.x;
  const int wave = tid >> 5;
  const int lane = tid & 31;
  float* G = sG[wave];

  // Current pose (rows 0..2 of 4x4, row-major) and intrinsics.
  const float R00 = pose[0], R01 = pose[1], R02 = pose[2],  tx = pose[3];
  const float R10 = pose[4], R11 = pose[5], R12 = pose[6],  ty = pose[7];
  const float R20 = pose[8], R21 = pose[9], R22 = pose[10], tz = pose[11];
  const float fx = Kg[0], cx = Kg[2], fy = Kg[4], cy = Kg[5];

  v8f acc = {};  // 16x16 f32 accumulator; top-left 8x8 holds G^T G

  const int wstride = (int)gridDim.x * 8;
  const int wid     = (int)blockIdx.x * 8 + wave;

  for (int base = wid * 32; base < NPIX; base += wstride * 32) {
    const int i  = base + lane;
    const int ic = (i < NPIX) ? i : (NPIX - 1);

    const float vx0 = v0g[3 * ic + 0], vy0 = v0g[3 * ic + 1], vz0 = v0g[3 * ic + 2];
    const float nx0 = n0g[3 * ic + 0], ny0 = n0g[3 * ic + 1], nz0 = n0g[3 * ic + 2];

    const float px = R00 * vx0 + R01 * vy0 + R02 * vz0 + tx;
    const float py = R10 * vx0 + R11 * vy0 + R12 * vz0 + ty;
    const float pz = R20 * vx0 + R21 * vy0 + R22 * vz0 + tz;
    const float n0x = R00 * nx0 + R01 * ny0 + R02 * nz0;
    const float n0y = R10 * nx0 + R11 * ny0 + R12 * nz0;
    const float n0z = R20 * nx0 + R21 * ny0 + R22 * nz0;

    const float invz = 1.0f / pz;
    const float u = px * invz * fx + cx;
    const float v = py * invz * fy + cy;
    const bool inview = (u > 0.f) & (u < (float)(W_IMG - 1)) &
                        (v > 0.f) & (v < (float)(H_IMG - 1));

    // round-half-even then clip (matches jnp.round + clip); fmin/fmax absorb NaN
    const float ur = fminf(fmaxf(rintf(u), 0.f), (float)(W_IMG - 1));
    const float vr = fminf(fmaxf(rintf(v), 0.f), (float)(H_IMG - 1));
    const int idx = (int)vr * W_IMG + (int)ur;

    const float rvx = v1g[3 * idx + 0], rvy = v1g[3 * idx + 1], rvz = v1g[3 * idx + 2];
    const float rnx = n1g[3 * idx + 0], rny = n1g[3 * idx + 1], rnz = n1g[3 * idx + 2];

    const float dx = px - rvx, dy = py - rvy, dz = pz - rvz;
    const bool mask0 = vz0 > 0.f;
    const bool mask1 = rvz > 0.f;
    const bool normal_ok = (n0x * rnx + n0y * rny + n0z * rnz) > NORM_THR;
    const bool occ = (!inview) | ((dx * dx + dy * dy + dz * dz) > DIST_THR2);
    const bool valid = (!occ) & mask0 & mask1 & normal_ok & (i < NPIX);
    const float m = valid ? 1.f : 0.f;

    const float res = rnx * dx + rny * dy + rnz * dz;
    G[lane * 8 + 0] = m * (py * rnz - pz * rny);
    G[lane * 8 + 1] = m * (pz * rnx - px * rnz);
    G[lane * 8 + 2] = m * (px * rny - py * rnx);
    G[lane * 8 + 3] = m * rnx;
    G[lane * 8 + 4] = m * rny;
    G[lane * 8 + 5] = m * rnz;
    G[lane * 8 + 6] = m * res;
    G[lane * 8 + 7] = m;

    // A layout (16x4 f32): lanes 0-15 row M=lane, v0=K0,v1=K1; lanes 16-31 K2/K3.
    // B layout (4x16 f32): identical per-lane values for the Gram product.
    const int comp = lane & 15;
    const int hsel = (lane >= 16) ? 2 : 0;
#pragma unroll
    for (int k = 0; k < 8; ++k) {
      const int p0 = 4 * k + hsel;
      v2f A;
      A[0] = (comp < 8) ? G[(p0 + 0) * 8 + comp] : 0.f;
      A[1] = (comp < 8) ? G[(p0 + 1) * 8 + comp] : 0.f;
      acc = __builtin_amdgcn_wmma_f32_16x16x4_f32(false, A, false, A,
                                                  (short)0, acc, false, false);
    }
  }

  // C layout: VGPR j, lanes 0-15 -> (M=j, N=lane). Useful 8x8 lives in lanes 0-7.
  if (lane < 8) {
#pragma unroll
    for (int j = 0; j < 8; ++j) sAcc[wave][j * 8 + lane] = acc[j];
  }
  __syncthreads();
  if (tid < 64) {
    float s = 0.f;
#pragma unroll
    for (int w = 0; w < 8; ++w) s += sAcc[w][tid];
    partials[(int)blockIdx.x * 64 + tid] = s;
  }
}

// ---------------------------------------------------------------------------
// Reduce partials, damp, solve 6x6, exp(se3), compose pose. Last iter writes
// pose + valid ratio to d_out.
// ---------------------------------------------------------------------------
__global__ void __launch_bounds__(64) icp_solve(
    const float* __restrict__ partials, int nb,
    float* __restrict__ pose_ws, float* __restrict__ d_out, int last) {
  __shared__ float A8[64];
  const int t = threadIdx.x;
  float s = 0.f;
  for (int b = 0; b < nb; ++b) s += partials[b * 64 + t];
  A8[t] = s;
  __syncthreads();
  if (t == 0) {
    // Augmented system [H | JtR]
    float M[6][7];
    for (int r = 0; r < 6; ++r) {
      for (int c = 0; c < 6; ++c) M[r][c] = A8[r * 8 + c];
      M[r][6] = A8[r * 8 + 6];
    }
    float tr = 0.f;
    for (int i = 0; i < 6; ++i) tr += M[i][i];
    for (int i = 0; i < 6; ++i) M[i][i] += tr * DAMPING;

    // Gauss-Jordan with partial pivoting
    for (int col = 0; col < 6; ++col) {
      int piv = col;
      float best = fabsf(M[col][col]);
      for (int r = col + 1; r < 6; ++r) {
        const float a = fabsf(M[r][col]);
        if (a > best) { best = a; piv = r; }
      }
      if (piv != col) {
        for (int c = 0; c < 7; ++c) {
          const float tmp = M[col][c]; M[col][c] = M[piv][c]; M[piv][c] = tmp;
        }
      }
      const float inv = 1.f / M[col][col];
      for (int c = col; c < 7; ++c) M[col][c] *= inv;
      for (int r = 0; r < 6; ++r) {
        if (r == col) continue;
        const float f = M[r][col];
        for (int c = col; c < 7; ++c) M[r][c] -= f * M[col][c];
      }
    }
    float xi[6];
    for (int r = 0; r < 6; ++r) xi[r] = -M[r][6];

    // exp_se3 (matches reference incl. eps handling)
    const float w0 = xi[0], w1 = xi[1], w2 = xi[2];
    const float vx = xi[3], vy = xi[4], vz = xi[5];
    const float theta = sqrtf(w0 * w0 + w1 * w1 + w2 * w2);
    const float ts = fmaxf(theta, 1e-8f);
    const float st = sinf(ts), ct = cosf(ts);
    const float cA = st / ts;
    const float cB = (1.f - ct) / (ts * ts);
    const float cC = (ts - st) / (ts * ts * ts);
    const float wh[3][3] = {{0.f, -w2, w1}, {w2, 0.f, -w0}, {-w1, w0, 0.f}};
    float wh2[3][3];
    for (int i = 0; i < 3; ++i)
      for (int j = 0; j < 3; ++j) {
        float a = 0.f;
        for (int k = 0; k < 3; ++k) a += wh[i][k] * wh[k][j];
        wh2[i][j] = a;
      }
    const bool small = theta <= 1e-8f;
    float E[3][3], Jm[3][3];
    for (int i = 0; i < 3; ++i)
      for (int j = 0; j < 3; ++j) {
        const float id = (i == j) ? 1.f : 0.f;
        E[i][j]  = small ? id : (id + cA * wh[i][j] + cB * wh2[i][j]);
        Jm[i][j] = small ? id : (id + cB * wh[i][j] + cC * wh2[i][j]);
      }
    const float t0 = Jm[0][0] * vx + Jm[0][1] * vy + Jm[0][2] * vz;
    const float t1 = Jm[1][0] * vx + Jm[1][1] * vy + Jm[1][2] * vz;
    const float t2 = Jm[2][0] * vx + Jm[2][1] * vy + Jm[2][2] * vz;

    // pose_new = [E t; 0 1] @ pose_old
    float P[16];
    for (int i = 0; i < 16; ++i) P[i] = pose_ws[i];
    float Pn[16];
    for (int r = 0; r < 3; ++r) {
      const float tr_ = (r == 0) ? t0 : (r == 1) ? t1 : t2;
      for (int c = 0; c < 4; ++c)
        Pn[r * 4 + c] = E[r][0] * P[0 + c] + E[r][1] * P[4 + c] +
                        E[r][2] * P[8 + c] + tr_ * P[12 + c];
    }
    Pn[12] = P[12]; Pn[13] = P[13]; Pn[14] = P[14]; Pn[15] = P[15];
    for (int i = 0; i < 16; ++i) pose_ws[i] = Pn[i];

    if (last) {
      for (int i = 0; i < 16; ++i) d_out[i] = Pn[i];
      d_out[16] = A8[63] / (float)NPIX;  // G[7]=valid indicator -> (7,7)=count
    }
  }
}

// ---------------------------------------------------------------------------
extern "C" void kernel_launch(void* const* d_in, const int* in_sizes, int n_in,
                              void* d_out, int out_size, void* d_ws, size_t ws_size,
                              hipStream_t stream) {
  const float* pose_in = (const float*)d_in[0];
  const float* v0 = (const float*)d_in[1];
  const float* v1 = (const float*)d_in[2];
  const float* n0 = (const float*)d_in[3];
  const float* n1 = (const float*)d_in[4];
  const float* K  = (const float*)d_in[5];
  float* out = (float*)d_out;

  float* ws       = (float*)d_ws;
  float* pose_ws  = ws;       // 16 floats
  float* partials = ws + 16;  // nb * 64 floats

  int nb = 640;  // 640 blocks * 8 waves * 32 lanes -> exactly 8 trips over NPIX
  const long cap = ((long)(ws_size / 4) - 16) / 64;
  if (cap < (long)nb) nb = (cap < 1) ? 1 : (int)cap;

  icp_init_pose<<<1, 32, 0, stream>>>(pose_in, pose_ws);
  for (int it = 0; it < 3; ++it) {
    icp_reduce<<<nb, 256, 0, stream>>>(v0, v1, n0, n1, K, pose_ws, partials);
    icp_solve<<<1, 64, 0, stream>>>(partials, nb, pose_ws, out, (it == 2) ? 1 : 0);
  }
}